// _ES2_7430293422757
// MI455X (gfx1250) — compile-verified
//
#include <hip/hip_runtime.h>
#include <stdint.h>
#include <math.h>

// ---------------------------------------------------------------------------
// ES-RNN dual-seasonality Holt-Winters (B=1024 series, T=1024 steps)
//  Kernel 1: per-series serial scan, rings in LDS, y staged via CDNA5
//            async global->LDS double buffering (ASYNCcnt path).
//  Kernel 2: q = log(y/(s1*s2)), llev = log(level)  (d_ws scratch)
//  Kernel 3: window outputs = q - llev, vectorized float4 stores (~280MB)
// ---------------------------------------------------------------------------

#define SEAS1 24
#define SEAS2 168
#define TLEN  1024
#define BATCH 1024
#define ISZ   24
#define OSZ   48
#define NW    (TLEN - ISZ - OSZ + 1) /* 953 */

#define R1STR 25   /* ring1 LDS stride (floats), gcd(25,64)=1 -> no bank conflict */
#define R2STR 169  /* ring2 LDS stride (floats), gcd(169,64)=1 -> no bank conflict */
#define YSTR  65   /* y staging stride (floats), gcd(65,64)=1 -> no bank conflict */
#define CHUNK 32
#define NCH   (TLEN / CHUNK)

#if defined(__AMDGCN__) && __has_builtin(__builtin_amdgcn_global_load_async_to_lds_b32) && \
    __has_builtin(__builtin_amdgcn_s_wait_asynccnt)
#define HAVE_ASYNC 1
typedef int __attribute__((address_space(1)))* as1_ip;  // global i32*
typedef int __attribute__((address_space(3)))* as3_ip;  // LDS i32*
#else
#define HAVE_ASYNC 0
#endif

__device__ __forceinline__ float sigmoidf_(float x) {
  return 1.0f / (1.0f + expf(-x));
}

// ------------------------------ Kernel 1 -----------------------------------
__global__ __launch_bounds__(32) void es2_scan(
    const float* __restrict__ y, const float* __restrict__ lev_sms,
    const float* __restrict__ seas_sms1, const float* __restrict__ seas_sms2,
    const float* __restrict__ init_seas1, const float* __restrict__ init_seas2,
    const int* __restrict__ idxs,
    float* __restrict__ levels, float* __restrict__ s1o, float* __restrict__ s2o)
{
  __shared__ float ring1[32 * R1STR];
  __shared__ float ring2[32 * R2STR];
#if HAVE_ASYNC
  __shared__ float ybuf[2 * 32 * YSTR];
#endif
  const int tid = threadIdx.x;
  const int b   = blockIdx.x * 32 + tid;
  const int g   = idxs[b];

  const float lsm  = sigmoidf_(lev_sms[g]);
  const float s1sm = sigmoidf_(seas_sms1[g]);
  const float s2sm = sigmoidf_(seas_sms2[g]);

  float* r1 = &ring1[tid * R1STR];
  float* r2 = &ring2[tid * R2STR];
  float* s1row  = s1o + (size_t)b * (SEAS1 + TLEN);
  float* s2row  = s2o + (size_t)b * (SEAS2 + TLEN);
  float* levrow = levels + (size_t)b * TLEN;
  const float* yrow = y + (size_t)b * TLEN;

  // init seasonal rings (= exp(init_seas)); also emit the init part of output
  for (int j = 0; j < SEAS1; ++j) {
    float v = expf(init_seas1[(size_t)g * SEAS1 + j]);
    r1[j] = v; s1row[j] = v;
  }
  for (int j = 0; j < SEAS2; ++j) {
    float v = expf(init_seas2[(size_t)g * SEAS2 + j]);
    r2[j] = v; s2row[j] = v;
  }
  // reference: seas[:, S] = exp(init_seas[:,0]) (slot for t=0, never recomputed)
  s1row[SEAS1] = r1[0];
  s2row[SEAS2] = r2[0];

#if HAVE_ASYNC
  auto issue_chunk = [&](int c) {
    const float* gsrc = yrow + c * CHUNK;
    float* ldst = &ybuf[(c & 1) * (32 * YSTR) + tid * YSTR];
#pragma unroll
    for (int k = 0; k < CHUNK; ++k) {
      __builtin_amdgcn_global_load_async_to_lds_b32(
          (as1_ip)(uintptr_t)(gsrc + k),
          (as3_ip)(uint32_t)(uintptr_t)(ldst + k),
          0, 0);
    }
  };
  issue_chunk(0);
#endif

  float lev = 0.0f;
  for (int c = 0; c < NCH; ++c) {
#if HAVE_ASYNC
    // Double buffer: kick next chunk, wait for current. Async loads complete
    // in order, so asynccnt<=CHUNK guarantees chunk c has landed in LDS.
    if (c + 1 < NCH) { issue_chunk(c + 1); __builtin_amdgcn_s_wait_asynccnt(CHUNK); }
    else             { __builtin_amdgcn_s_wait_asynccnt(0); }
    const float* ybase = &ybuf[(c & 1) * (32 * YSTR) + tid * YSTR];
#else
    const float* ybase = yrow + c * CHUNK;
#endif
#pragma unroll 4
    for (int k = 0; k < CHUNK; ++k) {
      const int t = c * CHUNK + k;
      const float yt = ybase[k];
      if (t == 0) {
        lev = __fdividef(yt, r1[0] * r2[0]);
        levrow[0] = lev;
        continue;
      }
      const int p1 = t % SEAS1;
      const int p2 = t % SEAS2;
      const float sea1 = r1[p1];
      const float sea2 = r2[p2];
      // rr = y/(s1*s2); newlev = lev + lsm*(rr-lev)
      const float rr = __fdividef(yt, sea1 * sea2);
      const float newlev = fmaf(lsm, rr - lev, lev);
      // y/(newlev*s2) = s1*rr/newlev ; y/(newlev*s1) = s2*rr/newlev
      const float gq  = __fdividef(rr, newlev);
      const float gm1 = gq - 1.0f;
      const float new1 = fmaf(s1sm * sea1, gm1, sea1);
      const float new2 = fmaf(s2sm * sea2, gm1, sea2);
      r1[p1] = new1;
      r2[p2] = new2;
      levrow[t] = newlev;          // scattered across lanes, but sequential in t
      s1row[SEAS1 + t] = new1;     // -> lines stay dirty in L2, no amplification
      s2row[SEAS2 + t] = new2;
      lev = newlev;
    }
  }
}

// ------------------------------ Kernel 2 -----------------------------------
__global__ __launch_bounds__(256) void es2_logpre(
    const float* __restrict__ y, const float* __restrict__ levels,
    const float* __restrict__ s1o, const float* __restrict__ s2o,
    float* __restrict__ q, float* __restrict__ llev)
{
  const int i = blockIdx.x * blockDim.x + threadIdx.x;  // 0 .. B*T-1
  const int b = i >> 10;
  const int t = i & (TLEN - 1);
  const float s = s1o[b * (SEAS1 + TLEN) + t] * s2o[b * (SEAS2 + TLEN) + t];
  q[i]    = logf(__fdividef(y[i], s));
  llev[i] = logf(levels[i]);
}

// ------------------------------ Kernel 3 -----------------------------------
// windows_y_hat[w,b,i] = q[b][w+i] - llev[b][w+ISZ-1]    (nW,B,24)
__global__ __launch_bounds__(256) void es2_win_hat(
    const float* __restrict__ q, const float* __restrict__ llev,
    float4* __restrict__ out)
{
  const unsigned u = blockIdx.x * blockDim.x + threadIdx.x;  // NW*B*6
  if (u >= (unsigned)NW * BATCH * (ISZ / 4)) return;
  const unsigned i4 = (u % 6u) * 4u;
  const unsigned r  = u / 6u;
  const unsigned b  = r & (BATCH - 1);
  const unsigned w  = r >> 10;
  const float l = llev[b * TLEN + w + (ISZ - 1)];
  const unsigned base = b * TLEN + w + i4;
  float4 v;
  v.x = q[base + 0] - l; v.y = q[base + 1] - l;
  v.z = q[base + 2] - l; v.w = q[base + 3] - l;
  out[u] = v;
}

// windows_y[w,b,o] = q[b][w+ISZ+o] - llev[b][w+ISZ]      (nW,B,48)
__global__ __launch_bounds__(256) void es2_win_y(
    const float* __restrict__ q, const float* __restrict__ llev,
    float4* __restrict__ out)
{
  const unsigned u = blockIdx.x * blockDim.x + threadIdx.x;  // NW*B*12
  if (u >= (unsigned)NW * BATCH * (OSZ / 4)) return;
  const unsigned i4 = (u % 12u) * 4u;
  const unsigned r  = u / 12u;
  const unsigned b  = r & (BATCH - 1);
  const unsigned w  = r >> 10;
  const float l = llev[b * TLEN + w + ISZ];
  const unsigned base = b * TLEN + w + ISZ + i4;
  float4 v;
  v.x = q[base + 0] - l; v.y = q[base + 1] - l;
  v.z = q[base + 2] - l; v.w = q[base + 3] - l;
  out[u] = v;
}

// Fallback window kernels (no scratch): recompute logs directly.
__global__ __launch_bounds__(256) void es2_win_hat_direct(
    const float* __restrict__ y, const float* __restrict__ levels,
    const float* __restrict__ s1o, const float* __restrict__ s2o,
    float* __restrict__ out)
{
  const unsigned u = blockIdx.x * blockDim.x + threadIdx.x;  // NW*B*24
  if (u >= (unsigned)NW * BATCH * ISZ) return;
  const unsigned i = u % (unsigned)ISZ;
  const unsigned r = u / (unsigned)ISZ;
  const unsigned b = r & (BATCH - 1);
  const unsigned w = r >> 10;
  const unsigned j = w + i;
  const float den = s1o[b * (SEAS1 + TLEN) + j] * s2o[b * (SEAS2 + TLEN) + j] *
                    levels[b * TLEN + w + (ISZ - 1)];
  out[u] = logf(__fdividef(y[b * TLEN + j], den));
}

__global__ __launch_bounds__(256) void es2_win_y_direct(
    const float* __restrict__ y, const float* __restrict__ levels,
    const float* __restrict__ s1o, const float* __restrict__ s2o,
    float* __restrict__ out)
{
  const unsigned u = blockIdx.x * blockDim.x + threadIdx.x;  // NW*B*48
  if (u >= (unsigned)NW * BATCH * OSZ) return;
  const unsigned i = u % (unsigned)OSZ;
  const unsigned r = u / (unsigned)OSZ;
  const unsigned b = r & (BATCH - 1);
  const unsigned w = r >> 10;
  const unsigned j = w + ISZ + i;
  const float den = s1o[b * (SEAS1 + TLEN) + j] * s2o[b * (SEAS2 + TLEN) + j] *
                    levels[b * TLEN + w + ISZ];
  out[u] = logf(__fdividef(y[b * TLEN + j], den));
}

// ---------------------------------------------------------------------------
extern "C" void kernel_launch(void* const* d_in, const int* in_sizes, int n_in,
                              void* d_out, int out_size, void* d_ws, size_t ws_size,
                              hipStream_t stream) {
  (void)in_sizes; (void)n_in; (void)out_size;
  const float* y         = (const float*)d_in[0];
  const float* lev_sms   = (const float*)d_in[1];
  const float* seas_sms1 = (const float*)d_in[2];
  const float* seas_sms2 = (const float*)d_in[3];
  const float* init_s1   = (const float*)d_in[4];
  const float* init_s2   = (const float*)d_in[5];
  const int*   idxs      = (const int*)d_in[6];

  float* out = (float*)d_out;
  const size_t n_hat = (size_t)NW * BATCH * ISZ;          // 23,420,928
  const size_t n_y   = (size_t)NW * BATCH * OSZ;          // 46,841,856
  float* out_hat = out;
  float* out_y   = out + n_hat;
  float* levels  = out_y + n_y;
  float* s1o     = levels + (size_t)BATCH * TLEN;
  float* s2o     = s1o + (size_t)BATCH * (SEAS1 + TLEN);

  // Phase 1: serial scan, 1 thread per series, 32 waves total.
  es2_scan<<<BATCH / 32, 32, 0, stream>>>(y, lev_sms, seas_sms1, seas_sms2,
                                          init_s1, init_s2, idxs,
                                          levels, s1o, s2o);

  const size_t need = 2ull * BATCH * TLEN * sizeof(float);  // 8 MB
  if (ws_size >= need) {
    float* q    = (float*)d_ws;
    float* llev = q + (size_t)BATCH * TLEN;
    es2_logpre<<<(BATCH * TLEN) / 256, 256, 0, stream>>>(y, levels, s1o, s2o, q, llev);
    // exact divisibility: 953*1024*6/256 = 22872, 953*1024*12/256 = 45744
    es2_win_hat<<<(NW * BATCH * (ISZ / 4)) / 256, 256, 0, stream>>>(q, llev, (float4*)out_hat);
    es2_win_y  <<<(NW * BATCH * (OSZ / 4)) / 256, 256, 0, stream>>>(q, llev, (float4*)out_y);
  } else {
    es2_win_hat_direct<<<(NW * BATCH * ISZ) / 256, 256, 0, stream>>>(y, levels, s1o, s2o, out_hat);
    es2_win_y_direct  <<<(NW * BATCH * OSZ) / 256, 256, 0, stream>>>(y, levels, s1o, s2o, out_y);
  }
}